// BinomialLoss_4252017623385
// MI455X (gfx1250) — compile-verified
//
#include <hip/hip_runtime.h>
#include <hip/hip_bf16.h>
#include <math.h>
#include <stdint.h>

typedef float v2f __attribute__((ext_vector_type(2)));
typedef float v8f __attribute__((ext_vector_type(8)));

#define DDIM 512
#define ROWS 16
#define CK 64                    // columns per chunk
#define LSTRIDE 68               // padded floats per LDS row (bank-conflict-free)
#define MATF (ROWS * LSTRIDE)    // 1088 floats per matrix chunk
#define BUFF (2 * MATF)          // o1 chunk + o2 chunk = 2176 floats
#define WAVE_LDS_F (2 * BUFF)    // double-buffered: 4352 floats = 17408 B / wave
#define NCHUNK (DDIM / CK)       // 8

// ---- CDNA5 async copy: global -> LDS, 16B per lane, tracked by ASYNCcnt ----
__device__ __forceinline__ void async_b128(uint32_t lds_off, const float* g) {
  asm volatile("global_load_async_to_lds_b128 %0, %1, off"
               :: "v"(lds_off), "v"(g)
               : "memory");
}

// Issue one 16x64 chunk of both matrices into this wave's LDS buffer.
// Each of the 8 async instructions moves 2 full rows (32 lanes x 16B).
__device__ __forceinline__ void issue_chunk(const float* __restrict__ o1,
                                            const float* __restrict__ o2,
                                            size_t row0, int c0,
                                            const float* lbuf, int lane) {
  // make sure prior ds reads of this buffer have drained before DMA overwrites it
  asm volatile("s_wait_dscnt 0x0" ::: "memory");
  uint32_t lds_base = (uint32_t)(uintptr_t)lbuf;  // low 32 bits = LDS byte offset
  const int rl  = lane >> 4;          // 0 or 1: which of the 2 rows this lane feeds
  const int col = (lane & 15) << 2;   // float column (16B granule)
#pragma unroll
  for (int i = 0; i < 8; ++i) {
    const int r = 2 * i + rl;
    const uint32_t loff = lds_base + (uint32_t)((r * LSTRIDE + col) * 4);
    const float* g1 = o1 + (row0 + (size_t)r) * DDIM + (size_t)(c0 + col);
    const float* g2 = o2 + (row0 + (size_t)r) * DDIM + (size_t)(c0 + col);
    async_b128(loff, g1);
    async_b128(loff + (uint32_t)(MATF * 4), g2);
  }
}

__global__ void __launch_bounds__(256)
binomial_main(const float* __restrict__ o1, const float* __restrict__ o2,
              const int* __restrict__ target, float* __restrict__ partials,
              long long ntiles) {
  extern __shared__ float smem[];
  const int tid  = threadIdx.x;
  const int lane = tid & 31;
  const int wid  = tid >> 5;
  float* wbuf = smem + (size_t)wid * WAVE_LDS_F;

  const int wpb = blockDim.x >> 5;
  const long long gwave  = (long long)blockIdx.x * wpb + wid;
  const long long nwaves = (long long)gridDim.x * wpb;

  float pos_p = 0.f, neg_p = 0.f, cnt_p = 0.f;

  // A(16x4 f32) and B(4x16 f32) share the identical lane layout:
  // lane l, vgpr v  <->  element [row = l%16][k = k0 + v + 2*(l>>4)]
  const int m = lane & 15;
  const int h = (lane >> 4) * 2;

  for (long long tile = gwave; tile < ntiles; tile += nwaves) {
    const size_t row0 = (size_t)tile * ROWS;
    v8f acc_d = {0.f,0.f,0.f,0.f,0.f,0.f,0.f,0.f};  // diag -> dot(o1,o2)
    v8f acc_1 = {0.f,0.f,0.f,0.f,0.f,0.f,0.f,0.f};  // diag -> ||o1||^2
    v8f acc_2 = {0.f,0.f,0.f,0.f,0.f,0.f,0.f,0.f};  // diag -> ||o2||^2

    issue_chunk(o1, o2, row0, 0, wbuf, lane);
    for (int c = 0; c < NCHUNK; ++c) {
      const float* cur = wbuf + (c & 1) * BUFF;
      if (c + 1 < NCHUNK) {
        issue_chunk(o1, o2, row0, (c + 1) * CK, wbuf + ((c + 1) & 1) * BUFF, lane);
        // allow the 16 just-issued ops to stay in flight; wait for chunk c
        asm volatile("s_wait_asynccnt 16" ::: "memory");
      } else {
        asm volatile("s_wait_asynccnt 0" ::: "memory");
      }

      const float* p1 = cur + m * LSTRIDE + h;   // o1 operand base
      const float* p2 = p1 + MATF;               // o2 operand base
#pragma unroll
      for (int k = 0; k < CK; k += 4) {
        v2f a = *(const v2f*)(p1 + k);   // conflict-free ds_load_b64
        v2f b = *(const v2f*)(p2 + k);
        acc_d = __builtin_amdgcn_wmma_f32_16x16x4_f32(
            false, a, false, b, (short)0, acc_d, false, false);
        acc_1 = __builtin_amdgcn_wmma_f32_16x16x4_f32(
            false, a, false, a, (short)0, acc_1, false, false);
        acc_2 = __builtin_amdgcn_wmma_f32_16x16x4_f32(
            false, b, false, b, (short)0, acc_2, false, false);
      }
    }

    // Extract diagonal: C/D vgpr r holds (M=r, N=lane) on lanes 0-15 and
    // (M=r+8, N=lane-16) on lanes 16-31 -> (i,i) lives at {vgpr i, lane i}
    // for i<8 and {vgpr i-8, lane i+16} for i>=8. Distribute row i to lane i.
    float dot = 0.f, n1 = 0.f, n2 = 0.f;
#pragma unroll
    for (int r = 0; r < 8; ++r) {
      const float d0 = __shfl(acc_d[r], r, 32);
      const float d1 = __shfl(acc_d[r], r + 24, 32);
      const float a0 = __shfl(acc_1[r], r, 32);
      const float a1 = __shfl(acc_1[r], r + 24, 32);
      const float b0 = __shfl(acc_2[r], r, 32);
      const float b1 = __shfl(acc_2[r], r + 24, 32);
      if (lane == r)     { dot = d0; n1 = a0; n2 = b0; }
      if (lane == r + 8) { dot = d1; n1 = a1; n2 = b1; }
    }

    if (lane < 16) {
      const float denom = fmaxf(sqrtf(n1) * sqrtf(n2), 1e-6f);
      const float d = dot / denom;
      const float pos = 4.0f  * log1pf(expf(-0.5f * (d - 0.5f)));
      const float neg = 0.04f * log1pf(expf(50.0f * (d - 2.0f)));
      const int t = target[row0 + (size_t)lane];
      if (t == 1) { pos_p += pos; cnt_p += 1.0f; }
      else        { neg_p += neg; }
    }
  }

  // wave32 reduction (lanes 16-31 carry zeros)
#pragma unroll
  for (int off = 16; off > 0; off >>= 1) {
    pos_p += __shfl_xor(pos_p, off, 32);
    neg_p += __shfl_xor(neg_p, off, 32);
    cnt_p += __shfl_xor(cnt_p, off, 32);
  }
  if (lane == 0) {
    partials[gwave * 3 + 0] = pos_p;
    partials[gwave * 3 + 1] = neg_p;
    partials[gwave * 3 + 2] = cnt_p;
  }
}

__global__ void __launch_bounds__(256)
binomial_finalize(const float* __restrict__ partials, int nw,
                  float* __restrict__ out, float ntotal) {
  __shared__ float sp[256], sn[256], sc[256];
  const int tid = threadIdx.x;
  float p = 0.f, n = 0.f, c = 0.f;
  for (int i = tid; i < nw; i += 256) {   // fixed traversal -> deterministic
    p += partials[3 * i + 0];
    n += partials[3 * i + 1];
    c += partials[3 * i + 2];
  }
  sp[tid] = p; sn[tid] = n; sc[tid] = c;
  __syncthreads();
  for (int s = 128; s > 0; s >>= 1) {
    if (tid < s) {
      sp[tid] += sp[tid + s];
      sn[tid] += sn[tid + s];
      sc[tid] += sc[tid + s];
    }
    __syncthreads();
  }
  if (tid == 0) {
    const float np = sc[0];
    const float nn = ntotal - np;
    out[0] = sp[0] / fmaxf(np, 1.f) + sn[0] / fmaxf(nn, 1.f);
  }
}

extern "C" void kernel_launch(void* const* d_in, const int* in_sizes, int n_in,
                              void* d_out, int out_size, void* d_ws, size_t ws_size,
                              hipStream_t stream) {
  const float* o1     = (const float*)d_in[0];
  const float* o2     = (const float*)d_in[1];
  const int*   target = (const int*)d_in[2];
  float* out = (float*)d_out;

  const long long N      = (long long)in_sizes[2];  // rows
  const long long ntiles = N / ROWS;                // 8192 for N=131072

  float* partials = (float*)d_ws;

  const int blocks  = 512;
  const int threads = 256;                          // 8 waves/block
  const int nwaves  = blocks * (threads / 32);      // 4096 -> 48 KB workspace
  const size_t shmem = (size_t)(threads / 32) * WAVE_LDS_F * sizeof(float);

  binomial_main<<<blocks, threads, shmem, stream>>>(o1, o2, target, partials, ntiles);
  binomial_finalize<<<1, 256, 0, stream>>>(partials, nwaves, out, (float)N);
}